// HSTU_BSA_Layer_77979426226883
// MI455X (gfx1250) — compile-verified
//
#include <hip/hip_runtime.h>

#define BB  4
#define LL  1024
#define HH  8
#define DD  64
#define BS  32
#define SS  4
#define NBK (LL/BS)      // 32
#define TT  (BB*LL)      // 4096
#define HID (HH*DD)      // 512
#define G3H (3*HH)       // 24
#define EPSL 1e-5f

typedef float v2f __attribute__((ext_vector_type(2)));
typedef float v8f __attribute__((ext_vector_type(8)));

__device__ __forceinline__ float silu_f(float x) { return x / (1.f + __expf(-x)); }

// ---------------------------------------------------------------- block means
__global__ __launch_bounds__(256)
void cmp_mean_kernel(const float* __restrict__ k, const float* __restrict__ v,
                     float* __restrict__ k_cmp, float* __restrict__ v_cmp) {
    int idx = blockIdx.x * blockDim.x + threadIdx.x;     // B*NB*H*D
    if (idx >= BB*NBK*HH*DD) return;
    int d  = idx % DD;
    int h  = (idx / DD) % HH;
    int nb = (idx / (DD*HH)) % NBK;
    int b  = idx / (DD*HH*NBK);
    size_t base = ((size_t)(b*LL + nb*BS)*HH + h)*DD + d;
    float ks = 0.f, vs = 0.f;
    for (int c = 0; c < BS; ++c) {
        ks += k[base + (size_t)c*HH*DD];
        vs += v[base + (size_t)c*HH*DD];
    }
    k_cmp[idx] = ks * (1.f/BS);
    v_cmp[idx] = vs * (1.f/BS);
}

// ---------------------------------------------------------------- gate GEMM
__global__ __launch_bounds__(256)
void gates_kernel(const float* __restrict__ q, const float* __restrict__ Wg,
                  const float* __restrict__ bg, float* __restrict__ gates) {
    int idx = blockIdx.x * blockDim.x + threadIdx.x;     // T * 24
    if (idx >= TT*G3H) return;
    int j = idx % G3H, t = idx / G3H;
    const float* qr = q + (size_t)t*HID;
    float acc = bg[j];
    for (int i = 0; i < HID; ++i) acc += qr[i] * Wg[(size_t)i*G3H + j];
    gates[idx] = 1.f / (1.f + __expf(-acc));
}

// -------------------------------------------- compressed attention (WMMA f32)
__global__ __launch_bounds__(256)
void cmp_attn_kernel(const float* __restrict__ q, const float* __restrict__ k_cmp,
                     const float* __restrict__ v_cmp, const float* __restrict__ gates,
                     float* __restrict__ o_cmp, int* __restrict__ topk) {
    __shared__ float kcs[NBK][DD];          // 8 KB: K_cmp tile for this (b,h)
    __shared__ float vcs[NBK][DD];          // 8 KB
    __shared__ float sc[8][16][NBK];        // masked scores (for top-k)
    __shared__ float sp[8][16][NBK];        // masked silu(scores) (P matrix)

    int b = blockIdx.x / HH, h = blockIdx.x % HH;
    int wave = threadIdx.x >> 5;
    int lane = threadIdx.x & 31;
    int l0 = (blockIdx.y * 8 + wave) * 16;  // 16 query rows per wave

    for (int i = threadIdx.x; i < NBK*DD; i += 256) {
        int n = i / DD, d = i % DD;
        kcs[n][d] = k_cmp[((size_t)(b*NBK + n)*HH + h)*DD + d];
        vcs[n][d] = v_cmp[((size_t)(b*NBK + n)*HH + h)*DD + d];
    }
    __syncthreads();

    int Mn = lane & 15;   // A-row / B-col / C-col
    int kh = lane >> 4;   // K-half select for A/B operands, M-half for C

    // ---- scores: Q(16xD) x K_cmp^T(Dx32), two 16-wide N tiles ----
    v8f acc0 = {0.f,0.f,0.f,0.f,0.f,0.f,0.f,0.f};
    v8f acc1 = {0.f,0.f,0.f,0.f,0.f,0.f,0.f,0.f};
    const float* qrow = q + ((size_t)(b*LL + l0 + Mn)*HH + h)*DD;
    for (int kk = 0; kk < DD/4; ++kk) {
        int k0 = kk*4 + kh*2;
        v2f a  = { qrow[k0],        qrow[k0+1] };
        v2f b0 = { kcs[Mn][k0],     kcs[Mn][k0+1] };
        v2f b1 = { kcs[Mn+16][k0],  kcs[Mn+16][k0+1] };
        acc0 = __builtin_amdgcn_wmma_f32_16x16x4_f32(false, a, false, b0, (short)0, acc0, false, false);
        acc1 = __builtin_amdgcn_wmma_f32_16x16x4_f32(false, a, false, b1, (short)0, acc1, false, false);
    }

    // ---- scale, causal mask, silu -> LDS ----
    for (int r = 0; r < 8; ++r) {
        int Mrow = r + 8*kh;
        int l = l0 + Mrow;
        float s0 = acc0[r]*0.125f, s1 = acc1[r]*0.125f;
        bool c0 = (l / BS) >= Mn;
        bool c1 = (l / BS) >= (Mn + 16);
        sc[wave][Mrow][Mn]    = c0 ? s0 : -__builtin_inff();
        sc[wave][Mrow][Mn+16] = c1 ? s1 : -__builtin_inff();
        sp[wave][Mrow][Mn]    = c0 ? silu_f(s0) : 0.f;
        sp[wave][Mrow][Mn+16] = c1 ? silu_f(s1) : 0.f;
    }
    __syncthreads();

    // ---- top-4 per row (jax.lax.top_k tie-break: lowest index wins) ----
    if (lane < 16) {
        int l = l0 + lane;
        int sels[SS];
        for (int s = 0; s < SS; ++s) {
            int bn = -1; float bv = 0.f;
            for (int n = 0; n < NBK; ++n) {
                bool used = false;
                for (int t2 = 0; t2 < s; ++t2) if (sels[t2] == n) used = true;
                if (used) continue;
                float val = sc[wave][lane][n];
                if (bn < 0 || val > bv) { bv = val; bn = n; }
            }
            sels[s] = bn;
            topk[((size_t)(b*LL + l)*HH + h)*SS + s] = bn;
        }
    }

    // ---- o_cmp = P(16x32) x V_cmp(32x64), gated ----
    for (int nt = 0; nt < 4; ++nt) {
        v8f acc = {0.f,0.f,0.f,0.f,0.f,0.f,0.f,0.f};
        for (int kk = 0; kk < NBK/4; ++kk) {
            int k0 = kk*4 + kh*2;
            v2f a  = { sp[wave][Mn][k0],   sp[wave][Mn][k0+1] };
            v2f bb = { vcs[k0][nt*16+Mn],  vcs[k0+1][nt*16+Mn] };
            acc = __builtin_amdgcn_wmma_f32_16x16x4_f32(false, a, false, bb, (short)0, acc, false, false);
        }
        for (int r = 0; r < 8; ++r) {
            int Mrow = r + 8*kh;
            int l = l0 + Mrow;
            float g = gates[(size_t)(b*LL + l)*G3H + h];            // g_cmp
            o_cmp[((size_t)(b*LL + l)*HH + h)*DD + nt*16 + Mn] = acc[r] * g;
        }
    }
}

// --------------------------------------- selected (sparse) attention, per token
__global__ __launch_bounds__(128)
void slc_attn_kernel(const float* __restrict__ q, const float* __restrict__ k,
                     const float* __restrict__ v, const float* __restrict__ gates,
                     const int* __restrict__ topk, float* __restrict__ o_slc) {
    __shared__ float qs[DD];
    __shared__ float ps[SS*BS];
    __shared__ int   sel[SS];
    int th = blockIdx.x;            // t*H + h
    int h = th % HH, t = th / HH;
    int b = t / LL, l = t % LL;
    int tid = threadIdx.x;
    if (tid < SS) sel[tid] = topk[(size_t)th*SS + tid];
    if (tid < DD) qs[tid] = q[(size_t)th*DD + tid];
    __syncthreads();

    int s = tid >> 5, c = tid & 31;
    int pos = sel[s]*BS + c;
    const float* kr = k + ((size_t)(b*LL + pos)*HH + h)*DD;
    float scr = 0.f;
    #pragma unroll 8
    for (int d = 0; d < DD; ++d) scr += qs[d] * kr[d];
    scr *= 0.125f;
    ps[tid] = (pos <= l) ? silu_f(scr) : 0.f;
    __syncthreads();

    if (tid < DD) {
        float acc = 0.f;
        for (int c2 = 0; c2 < SS*BS; ++c2) {
            int pos2 = sel[c2 >> 5]*BS + (c2 & 31);
            acc += ps[c2] * v[((size_t)(b*LL + pos2)*HH + h)*DD + tid];
        }
        float g = gates[(size_t)t*G3H + HH + h];                    // g_slc
        o_slc[(size_t)th*DD + tid] = acc * g;
    }
}

// ---------------------------------------------------------------- LN + combine
__global__ __launch_bounds__(256)
void ln_combine_kernel(const float* __restrict__ o_cmp, const float* __restrict__ o_slc,
                       const float* __restrict__ u, float* __restrict__ out) {
    int wave = threadIdx.x >> 5, lane = threadIdx.x & 31;
    int t = blockIdx.x * 8 + wave;
    const float* oc = o_cmp + (size_t)t*HID;
    const float* os = o_slc + (size_t)t*HID;
    float sc = 0.f, sc2 = 0.f, ss = 0.f, ss2 = 0.f;
    for (int i = lane; i < HID; i += 32) {
        float a = oc[i]; sc += a; sc2 += a*a;
        float b = os[i]; ss += b; ss2 += b*b;
    }
    for (int off = 16; off > 0; off >>= 1) {
        sc  += __shfl_xor(sc,  off, 32);
        sc2 += __shfl_xor(sc2, off, 32);
        ss  += __shfl_xor(ss,  off, 32);
        ss2 += __shfl_xor(ss2, off, 32);
    }
    float mc = sc  * (1.f/HID), vc = sc2 * (1.f/HID) - mc*mc;
    float ms = ss  * (1.f/HID), vs = ss2 * (1.f/HID) - ms*ms;
    float ic = rsqrtf(vc + EPSL), is = rsqrtf(vs + EPSL);
    for (int i = lane; i < HID; i += 32) {
        out[(size_t)t*HID + i] = ((oc[i]-mc)*ic + (os[i]-ms)*is) * u[(size_t)t*HID + i];
    }
}

// ----------------------------------------------------------------------------
extern "C" void kernel_launch(void* const* d_in, const int* in_sizes, int n_in,
                              void* d_out, int out_size, void* d_ws, size_t ws_size,
                              hipStream_t stream) {
    const float* q  = (const float*)d_in[0];
    const float* k  = (const float*)d_in[1];
    const float* v  = (const float*)d_in[2];
    const float* u  = (const float*)d_in[3];
    const float* Wg = (const float*)d_in[4];
    const float* bg = (const float*)d_in[5];
    // d_in[6] = x_offsets (unused: fixed, uniform B*L layout)

    float* ws    = (float*)d_ws;
    float* gates = ws;                               // T*24
    float* k_cmp = gates + (size_t)TT*G3H;           // B*NB*H*D
    float* v_cmp = k_cmp + (size_t)BB*NBK*HH*DD;
    float* o_cmp = v_cmp + (size_t)BB*NBK*HH*DD;     // T*HID
    float* o_slc = o_cmp + (size_t)TT*HID;           // T*HID
    int*   topk  = (int*)(o_slc + (size_t)TT*HID);   // T*H*S ints

    cmp_mean_kernel<<<(BB*NBK*HH*DD + 255)/256, 256, 0, stream>>>(k, v, k_cmp, v_cmp);
    gates_kernel<<<(TT*G3H + 255)/256, 256, 0, stream>>>(q, Wg, bg, gates);
    cmp_attn_kernel<<<dim3(BB*HH, LL/128), 256, 0, stream>>>(q, k_cmp, v_cmp, gates, o_cmp, topk);
    slc_attn_kernel<<<TT*HH, 128, 0, stream>>>(q, k, v, gates, topk, o_slc);
    ln_combine_kernel<<<TT/8, 256, 0, stream>>>(o_cmp, o_slc, u, (float*)d_out);
}